// AttentionMechanism_54116587930138
// MI455X (gfx1250) — compile-verified
//
#include <hip/hip_runtime.h>
#include <math.h>

// Problem dims (fixed by the reference)
#define S_DIM 2048
#define B_DIM 32
#define H_DIM 1024

typedef float v2f __attribute__((ext_vector_type(2)));
typedef float v8f __attribute__((ext_vector_type(8)));

#if __has_builtin(__builtin_amdgcn_wmma_f32_16x16x4_f32)
#define HAVE_WMMA_F32X4 1
#else
#define HAVE_WMMA_F32X4 0
#endif

// ---------------------------------------------------------------------------
// Kernel 1: q[b,h] = sum_k dec[b,k] * W[k,h]   (M=32, N=1024, K=1024, all f32)
// One wave (32 threads) per 16x16 output tile. grid = (N/16, M/16) = (64, 2).
// Uses V_WMMA_F32_16X16X4_F32, K-loop of 256 steps.
//
// f32 A 16x4 layout (ISA 7.12.2): lanes 0-15 -> M=lane, VGPR0=K0, VGPR1=K1;
//                                 lanes 16-31 -> M=lane-16, VGPR0=K2, VGPR1=K3.
// B 4x16 mirrors it: lanes 0-15 -> N=lane, rows K0/K1; lanes 16-31 -> rows K2/K3.
// C/D 16x16: VGPR i holds M=i (lanes 0-15) and M=i+8 (lanes 16-31), N = lane%16.
// ---------------------------------------------------------------------------
__global__ __launch_bounds__(32) void proj_gemm_kernel(
    const float* __restrict__ dec,   // [B,H]
    const float* __restrict__ W,     // [H,H]  W[k,h]
    float* __restrict__ q)           // [B,H]
{
    const int n0 = blockIdx.x * 16;
    const int m0 = blockIdx.y * 16;
    const int lane = threadIdx.x;
    const int half = lane >> 4;      // 0: K pair {0,1}, 1: K pair {2,3}
    const int idx  = lane & 15;      // M (for A) / N (for B) within tile

#if HAVE_WMMA_F32X4
    v8f acc = {};
    const float* arow = dec + (size_t)(m0 + idx) * H_DIM;
    for (int kk = 0; kk < H_DIM; kk += 4) {
        const int ka = kk + 2 * half;
        v2f a, b;
        a.x = arow[ka];
        a.y = arow[ka + 1];
        b.x = W[(size_t)ka * H_DIM + n0 + idx];
        b.y = W[(size_t)(ka + 1) * H_DIM + n0 + idx];
        acc = __builtin_amdgcn_wmma_f32_16x16x4_f32(
            /*neg_a=*/false, a, /*neg_b=*/false, b,
            /*c_mod=*/(short)0, acc, /*reuse_a=*/false, /*reuse_b=*/false);
    }
    float* qout = q + (size_t)(m0 + half * 8) * H_DIM + n0 + idx;
#pragma unroll
    for (int i = 0; i < 8; ++i) qout[(size_t)i * H_DIM] = acc[i];
#else
    // Scalar fallback with identical output mapping (8 rows per lane).
    float accs[8];
#pragma unroll
    for (int i = 0; i < 8; ++i) accs[i] = 0.0f;
    for (int k = 0; k < H_DIM; ++k) {
        const float wv = W[(size_t)k * H_DIM + n0 + idx];
#pragma unroll
        for (int i = 0; i < 8; ++i)
            accs[i] = fmaf(dec[(size_t)(m0 + half * 8 + i) * H_DIM + k], wv, accs[i]);
    }
    float* qout = q + (size_t)(m0 + half * 8) * H_DIM + n0 + idx;
#pragma unroll
    for (int i = 0; i < 8; ++i) qout[(size_t)i * H_DIM] = accs[i];
#endif
}

// ---------------------------------------------------------------------------
// Kernel 2: energies[b,s] = sum_h enc[s,b,h] * q[b,h]
// Bandwidth-bound: streams 268 MB of enc once with float4 (global_load_b128).
// grid = (B, S/8); 256 threads = 8 waves; one (s,b) row per wave.
// q[b,:] (4 KB) staged in LDS once per block, reused by all 8 waves.
// ---------------------------------------------------------------------------
__global__ __launch_bounds__(256) void energies_kernel(
    const float* __restrict__ enc,   // [S,B,H]
    const float* __restrict__ q,     // [B,H]
    float* __restrict__ energies)    // [B,S]
{
    __shared__ __align__(16) float qs[H_DIM];
    const int b = blockIdx.x;
    const int sBase = blockIdx.y * 8;

    for (int i = threadIdx.x; i < H_DIM; i += 256)
        qs[i] = q[(size_t)b * H_DIM + i];
    __syncthreads();

    const int wave = threadIdx.x >> 5;
    const int lane = threadIdx.x & 31;
    const int s = sBase + wave;

    const float* erow = enc + ((size_t)s * B_DIM + b) * H_DIM;

    float4 acc = make_float4(0.f, 0.f, 0.f, 0.f);
#pragma unroll
    for (int i = 0; i < 8; ++i) {
        const int h = i * 128 + lane * 4;          // 32 lanes x float4 = 128 floats
        const float4 ev = *(const float4*)(erow + h);
        const float4 qv = *(const float4*)(qs + h);
        acc.x = fmaf(ev.x, qv.x, acc.x);
        acc.y = fmaf(ev.y, qv.y, acc.y);
        acc.z = fmaf(ev.z, qv.z, acc.z);
        acc.w = fmaf(ev.w, qv.w, acc.w);
    }
    float sum = (acc.x + acc.y) + (acc.z + acc.w);
#pragma unroll
    for (int m = 16; m >= 1; m >>= 1)
        sum += __shfl_xor(sum, m, 32);             // wave32 butterfly reduction
    if (lane == 0)
        energies[(size_t)b * S_DIM + s] = sum;
}

// ---------------------------------------------------------------------------
// Kernel 3: out[b, s] = softmax over s of energies[b, :]   (output is [B,1,S])
// One 256-thread block per b; 8 elements per thread.
// ---------------------------------------------------------------------------
__global__ __launch_bounds__(256) void softmax_kernel(
    const float* __restrict__ energies,  // [B,S]
    float* __restrict__ out)             // [B,S]
{
    __shared__ float red[256];
    const int b = blockIdx.x;
    const int t = threadIdx.x;
    const float* row = energies + (size_t)b * S_DIM;

    float v[8];
    float vmax = -INFINITY;
#pragma unroll
    for (int i = 0; i < 8; ++i) {
        v[i] = row[t + i * 256];
        vmax = fmaxf(vmax, v[i]);
    }
    red[t] = vmax;
    __syncthreads();
    for (int off = 128; off >= 1; off >>= 1) {
        if (t < off) red[t] = fmaxf(red[t], red[t + off]);
        __syncthreads();
    }
    vmax = red[0];
    __syncthreads();

    float lsum = 0.f;
#pragma unroll
    for (int i = 0; i < 8; ++i) {
        v[i] = expf(v[i] - vmax);
        lsum += v[i];
    }
    red[t] = lsum;
    __syncthreads();
    for (int off = 128; off >= 1; off >>= 1) {
        if (t < off) red[t] += red[t + off];
        __syncthreads();
    }
    const float inv = 1.0f / red[0];
#pragma unroll
    for (int i = 0; i < 8; ++i)
        out[(size_t)b * S_DIM + t + i * 256] = v[i] * inv;
}

// ---------------------------------------------------------------------------
// Launch. Inputs (setup_inputs order): decoder_hidden[B,H], encoder_outputs
// [S,B,H], W[H,H], b[H]. The bias contributes a per-b constant to the
// energies, which cancels exactly in softmax, so it is not needed.
// ---------------------------------------------------------------------------
extern "C" void kernel_launch(void* const* d_in, const int* in_sizes, int n_in,
                              void* d_out, int out_size, void* d_ws, size_t ws_size,
                              hipStream_t stream) {
    const float* dec = (const float*)d_in[0];
    const float* enc = (const float*)d_in[1];
    const float* W   = (const float*)d_in[2];
    (void)d_in;  (void)in_sizes; (void)n_in; (void)out_size; (void)ws_size;

    float* q        = (float*)d_ws;                    // B*H   = 32768 floats (128 KB)
    float* energies = q + (size_t)B_DIM * H_DIM;       // B*S   = 65536 floats (256 KB)
    float* out      = (float*)d_out;                   // B*1*S

    proj_gemm_kernel<<<dim3(H_DIM / 16, B_DIM / 16), 32, 0, stream>>>(dec, W, q);
    energies_kernel<<<dim3(B_DIM, S_DIM / 8), 256, 0, stream>>>(enc, q, energies);
    softmax_kernel<<<B_DIM, 256, 0, stream>>>(energies, out);
}